// R2B_flow_2516850835760
// MI455X (gfx1250) — compile-verified
//
#include <hip/hip_runtime.h>
#include <math.h>

// ---------------------------------------------------------------------------
// Shapes (fixed by the reference): N=1, C=64, H=480, W=360, K=32, Hr=64, Wr=2048
// ---------------------------------------------------------------------------
#define CH     64
#define H_     480
#define W_     360
#define HW     (H_ * W_)          // 172800
#define KS     32
#define HR     64
#define WR     2048
#define HRWR   (HR * WR)          // 131072
#define EPS    1e-5f

typedef float v2f __attribute__((ext_vector_type(2)));
typedef float v8f __attribute__((ext_vector_type(8)));

// ---------------------------------------------------------------------------
// Kernel 0a: range_fea [C,Hr,Wr] -> NHWC [Hr*Wr, C]  (gather source, L2-resident)
// ---------------------------------------------------------------------------
__global__ __launch_bounds__(256) void k_nhwc(const float* __restrict__ rin,
                                              float* __restrict__ nhwc) {
  int tid = blockIdx.x * 256 + threadIdx.x;          // over C*Hr*Wr
  if (tid >= CH * HRWR) return;
  int c   = tid >> 17;                               // / HRWR
  int pix = tid & (HRWR - 1);
  nhwc[pix * CH + c] = rin[tid];
}

// ---------------------------------------------------------------------------
// Kernel 0b: polar [64,HW] -> channel-pair layout pair[cp][p] = {ch[2cp],ch[2cp+1]}
// ---------------------------------------------------------------------------
__global__ __launch_bounds__(256) void k_pairpack(const float* __restrict__ src,
                                                  v2f* __restrict__ dst) {
  int cp = blockIdx.x / (HW / 256);                  // 0..31
  int p  = (blockIdx.x % (HW / 256)) * 256 + threadIdx.x;
  v2f v;
  v.x = src[(2 * cp    ) * HW + p];
  v.y = src[(2 * cp + 1) * HW + p];
  dst[cp * HW + p] = v;
}

// ---------------------------------------------------------------------------
// Kernel 0c: weight prepack + BN fold.
//   K-order: k = (kh*3+kw)*IC + ic.  Wp[k/2][oc] = {W[k][oc], W[k+1][oc]}
//   bn[] = {scale1[64], bias1[64], scale2[64], bias2[64]}
// ---------------------------------------------------------------------------
__global__ __launch_bounds__(256) void k_prepack(
    const float* __restrict__ fw, const float* __restrict__ aw,
    const float* __restrict__ g1, const float* __restrict__ b1,
    const float* __restrict__ m1, const float* __restrict__ v1,
    const float* __restrict__ g2, const float* __restrict__ b2,
    const float* __restrict__ m2, const float* __restrict__ v2,
    v2f* __restrict__ Wp1, v2f* __restrict__ Wp2, float* __restrict__ bn) {
  int tid = blockIdx.x * 256 + threadIdx.x;
  if (tid < 36864) {                                  // fusion: 576 k-pairs * 64 oc
    int k2 = tid >> 6, oc = tid & 63;
    int k  = k2 * 2;
    int khw = k / 128, ic = k % 128;                  // ic even -> ic+1 < 128
    int kh = khw / 3, kw = khw % 3;
    v2f w;
    w.x = fw[((oc * 128 + ic    ) * 3 + kh) * 3 + kw];
    w.y = fw[((oc * 128 + ic + 1) * 3 + kh) * 3 + kw];
    Wp1[tid] = w;
  } else if (tid < 36864 + 18432) {                   // attn: 288 k-pairs * 64 oc
    int t = tid - 36864;
    int k2 = t >> 6, oc = t & 63;
    int k  = k2 * 2;
    int khw = k / 64, ic = k % 64;
    int kh = khw / 3, kw = khw % 3;
    v2f w;
    w.x = aw[((oc * 64 + ic    ) * 3 + kh) * 3 + kw];
    w.y = aw[((oc * 64 + ic + 1) * 3 + kh) * 3 + kw];
    Wp2[t] = w;
  } else if (tid < 36864 + 18432 + 128) {
    int i = tid - (36864 + 18432);
    if (i < 64) {
      float s = g1[i] * rsqrtf(v1[i] + EPS);
      bn[i]       = s;
      bn[64 + i]  = b1[i] - m1[i] * s;
    } else {
      int c = i - 64;
      float s = g2[c] * rsqrtf(v2[c] + EPS);
      bn[128 + c] = s;
      bn[192 + c] = b2[c] - m2[c] * s;
    }
  }
}

// ---------------------------------------------------------------------------
// Kernel 1: flow gather + bilinear + max over K=32 -> flow features in
// channel-pair layout.  16 lanes per output pixel, each lane owns 4 channels.
// Boundary handling: clamped addresses + masks folded into bilinear weights
// (identical math to the reference's  v * valid) -> no exec divergence.
// ---------------------------------------------------------------------------
__global__ __launch_bounds__(256) void k_gather(const float* __restrict__ flowm,
                                                const float* __restrict__ nhwc,
                                                v2f* __restrict__ flowpair) {
  int tid = blockIdx.x * 256 + threadIdx.x;
  int p   = tid >> 4;                 // pixel (y*W + x)
  int cg  = tid & 15;                 // channel group of 4
  if (p >= HW) return;
  int cg4 = cg * 4;
  const float2* fp = (const float2*)(flowm + (size_t)p * (KS * 2));

  float4 acc;
  acc.x = -INFINITY; acc.y = -INFINITY; acc.z = -INFINITY; acc.w = -INFINITY;

  for (int k = 0; k < KS; ++k) {
    float2 g = fp[k];                 // {gy, gx}
    float ix = ((g.y + 1.f) * (float)WR - 1.f) * 0.5f;
    float iy = ((g.x + 1.f) * (float)HR - 1.f) * 0.5f;
    float x0f = floorf(ix), y0f = floorf(iy);
    float wx = ix - x0f,    wy = iy - y0f;
    int x0 = (int)x0f, y0 = (int)y0f;
    int x1 = x0 + 1,   y1 = y0 + 1;
    // clamped addresses (always-valid loads)
    int x0c = min(max(x0, 0), WR - 1), x1c = min(max(x1, 0), WR - 1);
    int y0c = min(max(y0, 0), HR - 1), y1c = min(max(y1, 0), HR - 1);
    const float4* r0 = (const float4*)(nhwc + (y0c * WR) * CH + cg4);
    const float4* r1 = (const float4*)(nhwc + (y1c * WR) * CH + cg4);
    float4 g00 = r0[x0c * (CH / 4)];
    float4 g01 = r0[x1c * (CH / 4)];
    float4 g10 = r1[x0c * (CH / 4)];
    float4 g11 = r1[x1c * (CH / 4)];
    // validity masks folded into bilinear weights
    float vy0 = ((unsigned)y0 < (unsigned)HR) ? 1.f : 0.f;
    float vy1 = ((unsigned)y1 < (unsigned)HR) ? 1.f : 0.f;
    float vx0 = ((unsigned)x0 < (unsigned)WR) ? 1.f : 0.f;
    float vx1 = ((unsigned)x1 < (unsigned)WR) ? 1.f : 0.f;
    float w00 = (1.f - wy) * (1.f - wx) * vy0 * vx0;
    float w01 = (1.f - wy) * wx         * vy0 * vx1;
    float w10 = wy * (1.f - wx)         * vy1 * vx0;
    float w11 = wy * wx                 * vy1 * vx1;
    float vx_ = w00 * g00.x + w01 * g01.x + w10 * g10.x + w11 * g11.x;
    float vyv = w00 * g00.y + w01 * g01.y + w10 * g10.y + w11 * g11.y;
    float vz  = w00 * g00.z + w01 * g01.z + w10 * g10.z + w11 * g11.z;
    float vw  = w00 * g00.w + w01 * g01.w + w10 * g10.w + w11 * g11.w;
    acc.x = fmaxf(acc.x, vx_);
    acc.y = fmaxf(acc.y, vyv);
    acc.z = fmaxf(acc.z, vz);
    acc.w = fmaxf(acc.w, vw);
  }
  v2f lo, hi;
  lo.x = acc.x; lo.y = acc.y;
  hi.x = acc.z; hi.y = acc.w;
  flowpair[(2 * cg    ) * HW + p] = lo;   // channels 4cg, 4cg+1
  flowpair[(2 * cg + 1) * HW + p] = hi;   // channels 4cg+2, 4cg+3
}

// ---------------------------------------------------------------------------
// WMMA implicit-GEMM 3x3 conv core (V_WMMA_F32_16X16X4_F32, wave32).
// Per wave: D tile = 64 oc x 16 pixels = 4 accumulators.
//   A 16x4 : lane m(<16): {W[K0][ocb+m], W[K0+1][ocb+m]}, lane m+16: K0+2/K0+3
//   B 4x16 : lane n(<16): {in[K0][px n], in[K0+1][px n]}, lane n+16: K0+2/K0+3
//   D 16x16: VGPR r -> oc = ocb + r + 8*(lane>=16), pixel = lane&15
// With K-order k=(kh*3+kw)*IC+ic and channel-pair inputs, B is ONE b64 load.
// ---------------------------------------------------------------------------
__device__ __forceinline__ void conv_mac_chunk(v8f c[4], const v2f* __restrict__ wrow,
                                               v2f b, int n) {
#pragma unroll
  for (int a = 0; a < 4; ++a) {
    v2f av = wrow[a * 16 + n];
    c[a] = __builtin_amdgcn_wmma_f32_16x16x4_f32(false, av, false, b,
                                                 (short)0, c[a], false, false);
  }
}

// One (kh,kw) tap over IC channels of one paired source.
//   bp    = srcpair + khalf*HW + pin   (per-lane)
//   wrow0 = Wp + (K0base/2 + khalf)*64 (per-wave)
__device__ __forceinline__ void conv_tap(v8f c[4], const v2f* __restrict__ bp,
                                         const v2f* __restrict__ wrow, float mvh,
                                         int n, int nchunk) {
  for (int t = 0; t < nchunk; ++t) {
    v2f b = bp[(size_t)(2 * t) * HW];
    b.x *= mvh;
    b.y *= mvh;
    conv_mac_chunk(c, wrow + (size_t)(2 * t) * 64, b, n);
  }
}

// Conv1: input = concat(polar, flow) [128,H,W] (pair layout), K = 1152.
// Epilogue: BN1 + ReLU -> respair
__global__ __launch_bounds__(256) void k_conv1(const v2f* __restrict__ polarpair,
                                               const v2f* __restrict__ flowpair,
                                               const v2f* __restrict__ Wp,
                                               const float* __restrict__ bn,
                                               v2f* __restrict__ respair) {
  int wave  = threadIdx.x >> 5;
  int lane  = threadIdx.x & 31;
  int n     = lane & 15;
  int khalf = lane >> 4;
  int p     = (blockIdx.x * 8 + wave) * 16 + n;
  int y = p / W_, x = p % W_;

  v8f c[4];
#pragma unroll
  for (int a = 0; a < 4; ++a) c[a] = (v8f)0.f;

  for (int kh = 0; kh < 3; ++kh) {
    int yi = y + kh - 1;
    bool vh = (yi >= 0) && (yi < H_);
    float mvh = vh ? 1.f : 0.f;
    for (int kw = 0; kw < 3; ++kw) {
      int xi = x + kw - 1;
      xi += (xi < 0) ? W_ : 0;
      xi -= (xi >= W_) ? W_ : 0;
      int pin = vh ? (yi * W_ + xi) : 0;            // always-valid address
      int K0base = (kh * 3 + kw) * 128;
      const v2f* wrow = Wp + (size_t)((K0base >> 1) + khalf) * 64;
      // polar half (ic 0..63), then flow half (ic 64..127)
      conv_tap(c, polarpair + (size_t)khalf * HW + pin, wrow,            mvh, n, 16);
      conv_tap(c, flowpair  + (size_t)khalf * HW + pin, wrow + 32 * 64,  mvh, n, 16);
    }
  }
  // BN + ReLU epilogue, stored as channel pairs (oc even for even r)
#pragma unroll
  for (int a = 0; a < 4; ++a) {
#pragma unroll
    for (int r = 0; r < 8; r += 2) {
      int oc = a * 16 + r + 8 * khalf;
      v2f v;
      v.x = fmaxf(c[a][r    ] * bn[oc    ] + bn[64 + oc    ], 0.f);
      v.y = fmaxf(c[a][r + 1] * bn[oc + 1] + bn[64 + oc + 1], 0.f);
      respair[(oc >> 1) * HW + p] = v;
    }
  }
}

// Conv2: input = respair [64,H,W], K = 576.
// Epilogue: BN2 + channel softmax (cross half-wave via shfl_xor 16) +
//           out = polar + res * att
__global__ __launch_bounds__(256) void k_conv2(const v2f* __restrict__ respair,
                                               const v2f* __restrict__ polarpair,
                                               const v2f* __restrict__ Wp,
                                               const float* __restrict__ bn,
                                               float* __restrict__ out) {
  int wave  = threadIdx.x >> 5;
  int lane  = threadIdx.x & 31;
  int n     = lane & 15;
  int khalf = lane >> 4;
  int p     = (blockIdx.x * 8 + wave) * 16 + n;
  int y = p / W_, x = p % W_;

  v8f c[4];
#pragma unroll
  for (int a = 0; a < 4; ++a) c[a] = (v8f)0.f;

  for (int kh = 0; kh < 3; ++kh) {
    int yi = y + kh - 1;
    bool vh = (yi >= 0) && (yi < H_);
    float mvh = vh ? 1.f : 0.f;
    for (int kw = 0; kw < 3; ++kw) {
      int xi = x + kw - 1;
      xi += (xi < 0) ? W_ : 0;
      xi -= (xi >= W_) ? W_ : 0;
      int pin = vh ? (yi * W_ + xi) : 0;
      int K0base = (kh * 3 + kw) * 64;
      const v2f* wrow = Wp + (size_t)((K0base >> 1) + khalf) * 64;
      conv_tap(c, respair + (size_t)khalf * HW + pin, wrow, mvh, n, 16);
    }
  }

  // BN2, then softmax over 64 channels of pixel p (lanes n and n+16 pair up).
  float mx = -INFINITY;
#pragma unroll
  for (int a = 0; a < 4; ++a) {
#pragma unroll
    for (int r = 0; r < 8; ++r) {
      int oc = a * 16 + r + 8 * khalf;
      float z = c[a][r] * bn[128 + oc] + bn[192 + oc];
      c[a][r] = z;
      mx = fmaxf(mx, z);
    }
  }
  mx = fmaxf(mx, __shfl_xor(mx, 16, 32));
  float s = 0.f;
#pragma unroll
  for (int a = 0; a < 4; ++a) {
#pragma unroll
    for (int r = 0; r < 8; ++r) {
      float e = expf(c[a][r] - mx);
      c[a][r] = e;
      s += e;
    }
  }
  s += __shfl_xor(s, 16, 32);
  float inv = 1.f / s;
#pragma unroll
  for (int a = 0; a < 4; ++a) {
#pragma unroll
    for (int r = 0; r < 8; r += 2) {
      int oc = a * 16 + r + 8 * khalf;
      int cp = oc >> 1;
      v2f rv = respair[cp * HW + p];
      v2f pv = polarpair[cp * HW + p];
      out[ oc      * HW + p] = pv.x + rv.x * (c[a][r    ] * inv);
      out[(oc + 1) * HW + p] = pv.y + rv.y * (c[a][r + 1] * inv);
    }
  }
}

// ---------------------------------------------------------------------------
// Launch
// ---------------------------------------------------------------------------
extern "C" void kernel_launch(void* const* d_in, const int* in_sizes, int n_in,
                              void* d_out, int out_size, void* d_ws, size_t ws_size,
                              hipStream_t stream) {
  (void)in_sizes; (void)n_in; (void)out_size; (void)ws_size;
  const float* flowm  = (const float*)d_in[0];   // [1,480,360,32,2]
  const float* polar  = (const float*)d_in[1];   // [1,64,480,360]
  const float* rangef = (const float*)d_in[2];   // [1,64,64,2048]
  const float* fw     = (const float*)d_in[3];   // [64,128,3,3]
  const float* g1 = (const float*)d_in[4];
  const float* b1 = (const float*)d_in[5];
  const float* m1 = (const float*)d_in[6];
  const float* v1 = (const float*)d_in[7];
  const float* aw = (const float*)d_in[8];       // [64,64,3,3]
  const float* g2 = (const float*)d_in[9];
  const float* b2 = (const float*)d_in[10];
  const float* m2 = (const float*)d_in[11];
  const float* v2 = (const float*)d_in[12];
  float* out = (float*)d_out;                    // [1,64,480,360]

  char* ws = (char*)d_ws;
  size_t off = 0;
  float* nhwc      = (float*)(ws + off); off += (size_t)CH * HRWR * 4;   // 32 MB
  v2f*   flowpair  = (v2f*)  (ws + off); off += (size_t)(CH/2) * HW * 8; // 44 MB
  v2f*   respair   = (v2f*)  (ws + off); off += (size_t)(CH/2) * HW * 8; // 44 MB
  v2f*   polarpair = (v2f*)  (ws + off); off += (size_t)(CH/2) * HW * 8; // 44 MB
  v2f*   Wp1       = (v2f*)  (ws + off); off += (size_t)576 * 64 * 8;    // 288 KB
  v2f*   Wp2       = (v2f*)  (ws + off); off += (size_t)288 * 64 * 8;    // 144 KB
  float* bn        = (float*)(ws + off); off += 256 * 4;

  // 0a: NHWC repack of range features (L2-resident gather source)
  k_nhwc<<<(CH * HRWR + 255) / 256, 256, 0, stream>>>(rangef, nhwc);
  // 0b: polar -> channel-pair layout
  k_pairpack<<<(CH / 2) * (HW / 256), 256, 0, stream>>>(polar, polarpair);
  // 0c: weight prepack into WMMA K-pair layout + BN constant fold
  k_prepack<<<(36864 + 18432 + 128 + 255) / 256, 256, 0, stream>>>(
      fw, aw, g1, b1, m1, v1, g2, b2, m2, v2, Wp1, Wp2, bn);
  // 1: bilinear flow gather + max over K
  k_gather<<<(HW * 16) / 256, 256, 0, stream>>>(flowm, nhwc, flowpair);
  // 2: fusion conv (WMMA f32) + BN + ReLU
  k_conv1<<<HW / (16 * 8), 256, 0, stream>>>(polarpair, flowpair, Wp1, bn, respair);
  // 3: attn conv (WMMA f32) + BN + softmax + residual
  k_conv2<<<HW / (16 * 8), 256, 0, stream>>>(respair, polarpair, Wp2, bn, out);
}